// InteractionModule_27745488732352
// MI455X (gfx1250) — compile-verified
//
#include <hip/hip_runtime.h>
#include <hip/hip_bf16.h>

typedef __attribute__((ext_vector_type(2))) float v2f;
typedef __attribute__((ext_vector_type(8))) float v8f;

#define B_ 16
#define T_ 256
#define P_ 256
#define H_ 256
#define D_ 128
#define NF_ (B_ * T_)      // 4096 frames
#define BETA_ 0.25f

// ---------------------------------------------------------------------------
// f32 WMMA: D(16x16,f32) = A(16x4,f32) * B(4x16,f32) + C
// A frag: lane<16 -> row=lane, {K0,K1}; lane>=16 -> row=lane-16, {K2,K3}
// B frag: lane<16 -> col=lane, {K0,K1}; lane>=16 -> col=lane-16, {K2,K3}
// D frag: vgpr r, lane<16 -> (row r,   col lane); lane>=16 -> (row r+8, col lane-16)
// ---------------------------------------------------------------------------
__device__ __forceinline__ v8f wmma4(v2f a, v2f b, v8f c) {
  return __builtin_amdgcn_wmma_f32_16x16x4_f32(
      false, a, false, b, (short)0, c, false, false);
}

// Fast transcendentals: v_exp_f32 + v_rcp_f32 (single TRANS ops) instead of
// the ~10-op IEEE divide sequence / ocml tanh polynomial.
__device__ __forceinline__ float sigmoidf_(float x) {
  return __builtin_amdgcn_rcpf(1.0f + __expf(-x));
}
__device__ __forceinline__ float siluf_(float x) { return x * sigmoidf_(x); }
__device__ __forceinline__ float tanhf_(float x) {
  return 2.0f * sigmoidf_(2.0f * x) - 1.0f;
}

__device__ __forceinline__ unsigned fenc(float f) {
  unsigned u = __float_as_uint(f);
  return (u & 0x80000000u) ? ~u : (u | 0x80000000u);
}
__device__ __forceinline__ float fdec(unsigned u) {
  return (u & 0x80000000u) ? __uint_as_float(u & 0x7FFFFFFFu) : __uint_as_float(~u);
}

// K-pair loader: clean b64 when K%4==0, clamped address + cndmask otherwise
template <int K>
__device__ __forceinline__ v2f ldpair(const float* __restrict__ row, int ka) {
  if constexpr ((K & 3) == 0) {
    return *(const v2f*)(row + ka);       // 8B-aligned: ka even, even row stride
  } else {
    const int k0 = (ka     < K) ? ka     : (K - 1);  // clamp: always in-bounds
    const int k1 = (ka + 1 < K) ? ka + 1 : (K - 1);
    const float x0 = row[k0];
    const float x1 = row[k1];
    v2f r;
    r.x = (ka     < K) ? x0 : 0.0f;       // value select, no exec branching
    r.y = (ka + 1 < K) ? x1 : 0.0f;
    return r;
  }
}

// ---------------------------------------------------------------------------
// GEMM: Y[m, ycol0+n] = act(sum_k X[m,k] * W[n,k] + bias[n])
// W: [N, ldW] row-major (torch Linear). ACT: 0=none, 1=SiLU.
// block = 256 (8 waves). Wave computes MT vertical 16x16 tiles sharing one
// B fragment (MT independent accumulator chains).
// grid = ((N+127)/128, M/(16*MT))
// ---------------------------------------------------------------------------
template <int K, int ACT, int MT>
__global__ __launch_bounds__(256) void gemm_wmma(
    const float* __restrict__ X, int ldX,
    const float* __restrict__ W, int ldW,
    const float* __restrict__ bias,
    float* __restrict__ Y, int ldY, int ycol0,
    int M, int N) {
  const int w    = threadIdx.x >> 5;
  const int lane = threadIdx.x & 31;
  const int half = lane >> 4;
  const int lid  = lane & 15;
  const int n0 = blockIdx.x * 128 + w * 16;
  const int m0 = blockIdx.y * (16 * MT);
  if (n0 >= N || m0 >= M) return;

  const float* wrow = W + (size_t)(n0 + lid) * ldW;
  const float* xrow[MT];
#pragma unroll
  for (int mt = 0; mt < MT; ++mt)
    xrow[mt] = X + (size_t)(m0 + mt * 16 + lid) * ldX;

  v8f c[MT];
#pragma unroll
  for (int mt = 0; mt < MT; ++mt) c[mt] = (v8f){};

#pragma unroll 4
  for (int k0 = 0; k0 < K; k0 += 4) {
    const int ka = k0 + 2 * half;
    const v2f b = ldpair<K>(wrow, ka);
#pragma unroll
    for (int mt = 0; mt < MT; ++mt) {
      const v2f a = ldpair<K>(xrow[mt], ka);
      c[mt] = wmma4(a, b, c[mt]);
    }
  }

  const int col = n0 + lid;
  const float bv = bias[col];
#pragma unroll
  for (int mt = 0; mt < MT; ++mt) {
    float* yb = Y + (size_t)(m0 + mt * 16 + 8 * half) * ldY + ycol0 + col;
#pragma unroll
    for (int r = 0; r < 8; ++r) {
      float v = c[mt][r] + bv;
      if (ACT == 1) v = siluf_(v);
      yb[(size_t)r * ldY] = v;
    }
  }
}

// ---------------------------------------------------------------------------
// Row LayerNorm over 256 features. grid = rows, block = 256.
// ---------------------------------------------------------------------------
__global__ __launch_bounds__(256) void ln_kernel(
    const float* __restrict__ X, float* __restrict__ Y,
    const float* __restrict__ g, const float* __restrict__ b) {
  __shared__ float red[256];
  const int row = blockIdx.x, tid = threadIdx.x;
  const float x = X[(size_t)row * 256 + tid];
  red[tid] = x;
  __syncthreads();
  for (int s = 128; s > 0; s >>= 1) {
    if (tid < s) red[tid] += red[tid + s];
    __syncthreads();
  }
  const float mean = red[0] * (1.0f / 256.0f);
  __syncthreads();
  const float d = x - mean;
  red[tid] = d * d;
  __syncthreads();
  for (int s = 128; s > 0; s >>= 1) {
    if (tid < s) red[tid] += red[tid + s];
    __syncthreads();
  }
  const float var = red[0] * (1.0f / 256.0f);
  Y[(size_t)row * 256 + tid] = d * rsqrtf(var + 1e-5f) * g[tid] + b[tid];
}

// ---------------------------------------------------------------------------
// GRU recurrence (one layer). Single block, 512 threads = 16 waves.
// h [16,256] in LDS. Wave w owns hidden cols [16w,16w+16); per step computes
// r/z/n tiles hg = h @ Whh^T (3 WMMA chains, b64 weight loads, ds_b64 h loads)
// then the gate update in D-fragment layout.
// ---------------------------------------------------------------------------
__global__ __launch_bounds__(512) void gru_recur_kernel(
    const float* __restrict__ xW,   // [B,T,768]
    const float* __restrict__ Whh,  // [768,256]
    const float* __restrict__ bhh,  // [768]
    float* __restrict__ y,          // [B,T,256] or null
    float* __restrict__ hT,         // [16,256] or null
    int T) {
  __shared__ float hs[16][260];     // 260*4 % 8 == 0 -> rows stay 8B-aligned
  const int tid  = threadIdx.x;
  const int w    = tid >> 5;
  const int lane = tid & 31;
  const int half = lane >> 4;
  const int lid  = lane & 15;

  for (int i = tid; i < 16 * 260; i += 512) ((float*)hs)[i] = 0.0f;

  const int col = w * 16 + lid;
  const float* wr = Whh + (size_t)(col)*256;
  const float* wz = Whh + (size_t)(col + 256) * 256;
  const float* wn = Whh + (size_t)(col + 512) * 256;
  const float brb = bhh[col], bzb = bhh[col + 256], bnb = bhh[col + 512];

  for (int t = 0; t < T; ++t) {
    __syncthreads();  // h(t-1) writes visible before GEMM reads
    v8f cr = {}, cz = {}, cn = {};
#pragma unroll 4
    for (int k0 = 0; k0 < 256; k0 += 4) {
      const int ka = k0 + 2 * half;
      const v2f a  = *(const v2f*)&hs[lid][ka];
      const v2f br = *(const v2f*)(wr + ka);
      const v2f bz = *(const v2f*)(wz + ka);
      const v2f bn = *(const v2f*)(wn + ka);
      cr = wmma4(a, br, cr);
      cz = wmma4(a, bz, cz);
      cn = wmma4(a, bn, cn);
    }
    __syncthreads();  // all GEMM reads of h done before updating h
#pragma unroll
    for (int r = 0; r < 8; ++r) {
      const int bb = r + 8 * half;  // batch row
      const float* xw = xW + ((size_t)bb * T + t) * 768;
      const float rg = sigmoidf_(xw[col] + cr[r] + brb);
      const float zg = sigmoidf_(xw[col + 256] + cz[r] + bzb);
      const float ng = tanhf_(xw[col + 512] + rg * (cn[r] + bnb));
      const float hold = hs[bb][col];
      const float hnew = (1.0f - zg) * ng + zg * hold;
      hs[bb][col] = hnew;
      if (y) y[((size_t)bb * T + t) * 256 + col] = hnew;
    }
  }
  __syncthreads();
  if (hT)
    for (int i = tid; i < 16 * 256; i += 512) hT[i] = hs[i >> 8][i & 255];
}

// ---------------------------------------------------------------------------
// Fused point MLP + max-over-points. One block per (b,t) frame, 256 threads.
// layer1: [P,3]->[P,256] SiLU (WMMA, K=3 padded to 4 via clamped loader),
// 64-point LDS chunks; layer2: [64,256]@[256,256]^T SiLU (WMMA K=256, b64 /
// ds_b64 loads); column max via fragment reduce + shfl_xor(16) + atomicMax.
// ---------------------------------------------------------------------------
__global__ __launch_bounds__(256) void point_mlp_kernel(
    const float* __restrict__ pts,  // [NF, 256, 3]
    const float* __restrict__ W1, const float* __restrict__ b1,  // [256,3],[256]
    const float* __restrict__ W2, const float* __restrict__ b2,  // [256,256],[256]
    float* __restrict__ tok, int ldTok) {
  __shared__ float s1[64][260];
  __shared__ unsigned smax[256];
  const int f    = blockIdx.x;
  const int tid  = threadIdx.x;
  const int w    = tid >> 5;
  const int lane = tid & 31;
  const int half = lane >> 4;
  const int lid  = lane & 15;
  const float* pw = pts + (size_t)f * P_ * 3;

  if (tid < 256) smax[tid] = 0u;  // below fenc() of any finite value

  for (int chunk = 0; chunk < 4; ++chunk) {
    __syncthreads();  // s1 reuse barrier (and smax init on first pass)
    // ---- layer 1: 64 tiles (4 Mtiles x 16 Ntiles), 8 per wave, K=4 ----
#pragma unroll
    for (int i = 0; i < 8; ++i) {
      const int tile = w * 8 + i, mt = tile >> 4, nt = tile & 15;
      const int p0 = chunk * 64 + mt * 16, n0 = nt * 16;
      const int ka = 2 * half;
      const v2f a = ldpair<3>(pw + (size_t)(p0 + lid) * 3, ka);
      const v2f b = ldpair<3>(W1 + (size_t)(n0 + lid) * 3, ka);
      v8f c = {};
      c = wmma4(a, b, c);
      const int colb = n0 + lid;
      const float bv = b1[colb];
#pragma unroll
      for (int r = 0; r < 8; ++r)
        s1[mt * 16 + r + 8 * half][colb] = siluf_(c[r] + bv);
    }
    __syncthreads();
    // ---- layer 2 + column max, K=256 ----
#pragma unroll
    for (int i = 0; i < 8; ++i) {
      const int tile = w * 8 + i, mt = tile >> 4, nt = tile & 15;
      const int n0 = nt * 16;
      const float* wq = W2 + (size_t)(n0 + lid) * 256;
      v8f c = {};
#pragma unroll 4
      for (int k0 = 0; k0 < 256; k0 += 4) {
        const int ka = k0 + 2 * half;
        const v2f a = *(const v2f*)&s1[mt * 16 + lid][ka];
        const v2f b = *(const v2f*)(wq + ka);
        c = wmma4(a, b, c);
      }
      const int colb = n0 + lid;
      const float bv = b2[colb];
      float cmax = -3.4e38f;
#pragma unroll
      for (int r = 0; r < 8; ++r) cmax = fmaxf(cmax, siluf_(c[r] + bv));
      // lane l and l^16 hold the same column (rows 0-7 vs 8-15)
      cmax = fmaxf(cmax, __shfl_xor(cmax, 16, 32));
      if (half == 0) atomicMax(&smax[colb], fenc(cmax));
    }
  }
  __syncthreads();
  if (tid < 256) tok[(size_t)f * ldTok + tid] = fdec(smax[tid]);
}

// ---------------------------------------------------------------------------
// VQ: argmin over K=512 codes, z_q_st (== codebook[idx] forward value),
// vq_loss = (1+beta) * mean((z - z_q)^2). One block, 512 threads.
// ---------------------------------------------------------------------------
__global__ __launch_bounds__(512) void vq_kernel(
    const float* __restrict__ z,   // [16,128]
    const float* __restrict__ cb,  // [512,128]
    float* __restrict__ out_zq,    // [16,128]
    float* __restrict__ out_loss)  // [1]
{
  __shared__ float sd[512];
  __shared__ int si[512];
  __shared__ int idxs[16];
  const int tid = threadIdx.x;

  for (int b = 0; b < B_; ++b) {
    const float* zb = z + (size_t)b * D_;
    const float* ck = cb + (size_t)tid * D_;
    float d = 0.0f;
    for (int j = 0; j < D_; ++j) {
      const float t = zb[j] - ck[j];
      d += t * t;
    }
    sd[tid] = d;
    si[tid] = tid;
    __syncthreads();
    for (int s = 256; s > 0; s >>= 1) {
      if (tid < s) {
        const float do_ = sd[tid + s];
        const int io_ = si[tid + s];
        if (do_ < sd[tid] || (do_ == sd[tid] && io_ < si[tid])) {
          sd[tid] = do_;
          si[tid] = io_;
        }
      }
      __syncthreads();
    }
    if (tid == 0) idxs[b] = si[0];
    __syncthreads();
  }

  float acc = 0.0f;
  for (int e = tid; e < B_ * D_; e += 512) {
    const int b = e >> 7, j = e & 127;
    const float q = cb[(size_t)idxs[b] * D_ + j];
    out_zq[e] = q;
    const float diff = z[e] - q;
    acc += diff * diff;
  }
  sd[tid] = acc;
  __syncthreads();
  for (int s = 256; s > 0; s >>= 1) {
    if (tid < s) sd[tid] += sd[tid + s];
    __syncthreads();
  }
  if (tid == 0) out_loss[0] = (1.0f + BETA_) * sd[0] / (float)(B_ * D_);
}

// ---------------------------------------------------------------------------
extern "C" void kernel_launch(void* const* d_in, const int* in_sizes, int n_in,
                              void* d_out, int out_size, void* d_ws, size_t ws_size,
                              hipStream_t stream) {
  const float* obs_seq = (const float*)d_in[0];   // [16,256,159]
  const float* points  = (const float*)d_in[1];   // [16,256,256,3]
  const float* aux_seq = (const float*)d_in[2];   // [16,256,150]
  const float* obs_inW = (const float*)d_in[3];   // [256,159]
  const float* obs_inb = (const float*)d_in[4];
  const float* obs_lng = (const float*)d_in[5];
  const float* obs_lnb = (const float*)d_in[6];
  const float* oWih0 = (const float*)d_in[7];     // [768,256]
  const float* obih0 = (const float*)d_in[8];
  const float* oWhh0 = (const float*)d_in[9];
  const float* obhh0 = (const float*)d_in[10];
  const float* oWih1 = (const float*)d_in[11];
  const float* obih1 = (const float*)d_in[12];
  const float* oWhh1 = (const float*)d_in[13];
  const float* obhh1 = (const float*)d_in[14];
  const float* obs_outW = (const float*)d_in[15]; // [128,256]
  const float* obs_outb = (const float*)d_in[16];
  const float* m_p1W = (const float*)d_in[17];    // [256,3]
  const float* m_p1b = (const float*)d_in[18];
  const float* m_p2W = (const float*)d_in[19];    // [256,256]
  const float* m_p2b = (const float*)d_in[20];
  const float* m_a1W = (const float*)d_in[21];    // [256,150]
  const float* m_a1b = (const float*)d_in[22];
  const float* m_a2W = (const float*)d_in[23];    // [256,256]
  const float* m_a2b = (const float*)d_in[24];
  const float* m_fW  = (const float*)d_in[25];    // [256,512]
  const float* m_fb  = (const float*)d_in[26];
  const float* m_lng = (const float*)d_in[27];
  const float* m_lnb = (const float*)d_in[28];
  const float* mWih0 = (const float*)d_in[29];
  const float* mbih0 = (const float*)d_in[30];
  const float* mWhh0 = (const float*)d_in[31];
  const float* mbhh0 = (const float*)d_in[32];
  const float* mWih1 = (const float*)d_in[33];
  const float* mbih1 = (const float*)d_in[34];
  const float* mWhh1 = (const float*)d_in[35];
  const float* mbhh1 = (const float*)d_in[36];
  const float* m_outW = (const float*)d_in[37];   // [128,256]
  const float* m_outb = (const float*)d_in[38];
  const float* codebook = (const float*)d_in[39]; // [512,128]

  float* outF = (float*)d_out;  // [0:2048]=z_q_st, [2048:4096]=z_obs, [4096]=loss

  // workspace layout (floats)
  float* wsf    = (float*)d_ws;
  float* bufA   = wsf;                           // 4096*256  (pre-LN)
  float* bufH   = bufA + (size_t)NF_ * H_;       // 4096*256  (LN out)
  float* bufXW  = bufH + (size_t)NF_ * H_;       // 4096*768  (x @ Wih^T + bih)
  float* bufY   = bufXW + (size_t)NF_ * 3 * H_;  // 4096*256  (GRU layer0 out)
  float* tokcat = bufY + (size_t)NF_ * H_;       // 4096*512  [point_tok | aux]
  float* bufAux = tokcat + (size_t)NF_ * 2 * H_; // 4096*256
  float* hTb    = bufAux + (size_t)NF_ * H_;     // 16*256
  float* zmesh  = hTb + (size_t)B_ * H_;         // 16*128

  const dim3 blk(256);
  const dim3 gN256(2, NF_ / 64);   // N=256, M=4096, MT=4
  const dim3 gN768(6, NF_ / 64);   // N=768, M=4096, MT=4
  const dim3 gSmall(1, 1);         // M=16, N=128, MT=1

  // ===== obs branch =====
  gemm_wmma<159, 1, 4><<<gN256, blk, 0, stream>>>(obs_seq, 159, obs_inW, 159, obs_inb,
                                                  bufA, H_, 0, NF_, H_);
  ln_kernel<<<NF_, blk, 0, stream>>>(bufA, bufH, obs_lng, obs_lnb);
  gemm_wmma<256, 0, 4><<<gN768, blk, 0, stream>>>(bufH, H_, oWih0, H_, obih0,
                                                  bufXW, 3 * H_, 0, NF_, 3 * H_);
  gru_recur_kernel<<<1, 512, 0, stream>>>(bufXW, oWhh0, obhh0, bufY, nullptr, T_);
  gemm_wmma<256, 0, 4><<<gN768, blk, 0, stream>>>(bufY, H_, oWih1, H_, obih1,
                                                  bufXW, 3 * H_, 0, NF_, 3 * H_);
  gru_recur_kernel<<<1, 512, 0, stream>>>(bufXW, oWhh1, obhh1, nullptr, hTb, T_);
  gemm_wmma<256, 0, 1><<<gSmall, blk, 0, stream>>>(hTb, H_, obs_outW, H_, obs_outb,
                                                   outF + 2048, D_, 0, B_, D_);

  // ===== mesh branch =====
  point_mlp_kernel<<<NF_, blk, 0, stream>>>(points, m_p1W, m_p1b, m_p2W, m_p2b,
                                            tokcat, 2 * H_);
  gemm_wmma<150, 1, 4><<<gN256, blk, 0, stream>>>(aux_seq, 150, m_a1W, 150, m_a1b,
                                                  bufAux, H_, 0, NF_, H_);
  gemm_wmma<256, 1, 4><<<gN256, blk, 0, stream>>>(bufAux, H_, m_a2W, H_, m_a2b,
                                                  tokcat, 2 * H_, H_, NF_, H_);
  gemm_wmma<512, 1, 4><<<gN256, blk, 0, stream>>>(tokcat, 2 * H_, m_fW, 2 * H_, m_fb,
                                                  bufA, H_, 0, NF_, H_);
  ln_kernel<<<NF_, blk, 0, stream>>>(bufA, bufH, m_lng, m_lnb);
  gemm_wmma<256, 0, 4><<<gN768, blk, 0, stream>>>(bufH, H_, mWih0, H_, mbih0,
                                                  bufXW, 3 * H_, 0, NF_, 3 * H_);
  gru_recur_kernel<<<1, 512, 0, stream>>>(bufXW, mWhh0, mbhh0, bufY, nullptr, T_);
  gemm_wmma<256, 0, 4><<<gN768, blk, 0, stream>>>(bufY, H_, mWih1, H_, mbih1,
                                                  bufXW, 3 * H_, 0, NF_, 3 * H_);
  gru_recur_kernel<<<1, 512, 0, stream>>>(bufXW, mWhh1, mbhh1, nullptr, hTb, T_);
  gemm_wmma<256, 0, 1><<<gSmall, blk, 0, stream>>>(hTb, H_, m_outW, H_, m_outb,
                                                   zmesh, D_, 0, B_, D_);

  // ===== VQ =====
  vq_kernel<<<1, 512, 0, stream>>>(zmesh, codebook, outF, outF + 4096);

  (void)in_sizes; (void)n_in; (void)out_size; (void)ws_size;
}